// PopulationAttention_59682865545947
// MI455X (gfx1250) — compile-verified
//
#include <hip/hip_runtime.h>

// PopulationAttention (no softmax) on gfx1250:
//   out = Q @ (K^T @ (V*x)) * (1/sqrt(64))
// Reassociated to make it HBM-bound (~134MB @ 23.3TB/s) instead of compute
// bound; matrix work done with V_WMMA_F32_16X16X4_F32 (full f32 precision).

typedef float v2f __attribute__((ext_vector_type(2)));
typedef float v8f __attribute__((ext_vector_type(8)));

#define PA_N 2048
#define PA_D 64
#define PA_H 16
#define PA_BH 64   // B*H = 4*16

// ---------------------------------------------------------------------------
// Phase 1:  Mws[head] = 0.125f * K[head]^T @ (V[head] * x[b])      [64 x 64]
// grid = 64 heads, block = 256 threads (8 waves); each wave reduces 256 rows.
// ---------------------------------------------------------------------------
__global__ __launch_bounds__(256) void pa59_phase1(const float* __restrict__ K,
                                                   const float* __restrict__ V,
                                                   const float* __restrict__ x,
                                                   float* __restrict__ Mws) {
  __shared__ float Msh[PA_D * PA_D];   // 16 KB

  const int head = blockIdx.x;
  const int b    = head >> 4;          // H = 16
  const int tid  = threadIdx.x;
  const int lane = tid & 31;
  const int wave = tid >> 5;

  for (int i = tid; i < PA_D * PA_D; i += 256) Msh[i] = 0.0f;
  __syncthreads();

  const long hbase = (long)head * PA_N * PA_D;
  const float* __restrict__ Kh = K + hbase;
  const float* __restrict__ Vh = V + hbase;
  const float* __restrict__ xb = x + (long)b * PA_N;

  const int col  = lane & 15;               // N-dim lane within tile
  const int roff = (lane >= 16) ? 2 : 0;    // K-dim pair select (VGPR layout)

  v8f acc[4][4];
#pragma unroll
  for (int i = 0; i < 4; ++i)
#pragma unroll
    for (int j = 0; j < 4; ++j)
      acc[i][j] = (v8f){0.f, 0.f, 0.f, 0.f, 0.f, 0.f, 0.f, 0.f};

  const int r0 = wave * 256;
  for (int n0 = r0; n0 < r0 + 256; n0 += 4) {
    const int rA = n0 + roff;               // this lane's pair of N-rows
    const float xv0 = xb[rA];
    const float xv1 = xb[rA + 1];

    float a0[4], a1[4], b0[4], b1[4];
#pragma unroll
    for (int t = 0; t < 4; ++t) {
      const int c = t * 16 + col;
      a0[t] = Kh[(long)rA * PA_D + c];            // A = K^T : [d x n]
      a1[t] = Kh[(long)(rA + 1) * PA_D + c];
      b0[t] = Vh[(long)rA * PA_D + c] * xv0;      // B = V*x : [n x d]
      b1[t] = Vh[(long)(rA + 1) * PA_D + c] * xv1;
    }

#pragma unroll
    for (int ti = 0; ti < 4; ++ti) {
      v2f a;  a.x = a0[ti];  a.y = a1[ti];
#pragma unroll
      for (int tj = 0; tj < 4; ++tj) {
        v2f bb;  bb.x = b0[tj];  bb.y = b1[tj];
        acc[ti][tj] = __builtin_amdgcn_wmma_f32_16x16x4_f32(
            false, a, false, bb, (short)0, acc[ti][tj], false, false);
      }
    }
  }

  // Merge the 8 wave-partials via LDS float atomics (ds_add_f32).
  const int rowAdd = 8 * (lane >> 4);       // C/D layout: lanes 16-31 hold M+8
#pragma unroll
  for (int ti = 0; ti < 4; ++ti)
#pragma unroll
    for (int tj = 0; tj < 4; ++tj)
#pragma unroll
      for (int v = 0; v < 8; ++v) {
        const int row = ti * 16 + v + rowAdd;
        atomicAdd(&Msh[row * PA_D + tj * 16 + col], acc[ti][tj][v]);
      }
  __syncthreads();

  float* __restrict__ Mh = Mws + head * (PA_D * PA_D);
  for (int i = tid; i < PA_D * PA_D; i += 256)
    Mh[i] = Msh[i] * 0.125f;                // 1/sqrt(DIM_QK)
}

// ---------------------------------------------------------------------------
// Phase 2:  out[head] = Q[head] @ Mws[head]         [2048 x 64] @ [64 x 64]
// grid = 64 heads * 16 chunks, block = 256 threads; one wave per 16-row tile.
// ---------------------------------------------------------------------------
__global__ __launch_bounds__(256) void pa59_phase2(const float* __restrict__ Q,
                                                   const float* __restrict__ Mws,
                                                   float* __restrict__ out) {
  __shared__ float Msh[PA_D * PA_D];   // 16 KB, whole 64x64 M for this head

  const int head  = blockIdx.x >> 4;
  const int chunk = blockIdx.x & 15;
  const int tid   = threadIdx.x;
  const int lane  = tid & 31;
  const int wave  = tid >> 5;

  const float* __restrict__ Mh = Mws + head * (PA_D * PA_D);
  for (int i = tid; i < PA_D * PA_D; i += 256) Msh[i] = Mh[i];
  __syncthreads();

  const long hbase = (long)head * PA_N * PA_D;
  const float* __restrict__ Qh = Q + hbase;
  float* __restrict__ Oh       = out + hbase;

  const int col  = lane & 15;
  const int roff = (lane >= 16) ? 2 : 0;
  const int row0 = chunk * 128 + wave * 16;

  v8f acc[4];
#pragma unroll
  for (int j = 0; j < 4; ++j)
    acc[j] = (v8f){0.f, 0.f, 0.f, 0.f, 0.f, 0.f, 0.f, 0.f};

  // A-fragment rows: this lane always reads Q row (row0 + col); the two K
  // elements (k, k+1) are adjacent in memory -> coalesced float2 loads.
  const float* __restrict__ qrow = Qh + (long)(row0 + col) * PA_D;

  for (int k0 = 0; k0 < PA_D; k0 += 4) {
    v2f a = *reinterpret_cast<const v2f*>(qrow + k0 + roff);
    const int rB = k0 + roff;
#pragma unroll
    for (int tj = 0; tj < 4; ++tj) {
      v2f bb;
      bb.x = Msh[rB * PA_D + tj * 16 + col];
      bb.y = Msh[(rB + 1) * PA_D + tj * 16 + col];
      acc[tj] = __builtin_amdgcn_wmma_f32_16x16x4_f32(
          false, a, false, bb, (short)0, acc[tj], false, false);
    }
  }

  const int rowAdd = 8 * (lane >> 4);
#pragma unroll
  for (int tj = 0; tj < 4; ++tj)
#pragma unroll
    for (int v = 0; v < 8; ++v)
      Oh[(long)(row0 + v + rowAdd) * PA_D + tj * 16 + col] = acc[tj][v];
}

// ---------------------------------------------------------------------------
extern "C" void kernel_launch(void* const* d_in, const int* in_sizes, int n_in,
                              void* d_out, int out_size, void* d_ws, size_t ws_size,
                              hipStream_t stream) {
  (void)in_sizes; (void)n_in; (void)out_size; (void)ws_size;
  const float* Q = (const float*)d_in[0];
  const float* K = (const float*)d_in[1];
  const float* V = (const float*)d_in[2];
  const float* x = (const float*)d_in[3];
  float* out = (float*)d_out;
  float* Mws = (float*)d_ws;   // needs 64 heads * 64*64 * 4B = 1 MB

  pa59_phase1<<<PA_BH, 256, 0, stream>>>(K, V, x, Mws);
  pa59_phase2<<<PA_BH * 16, 256, 0, stream>>>(Q, Mws, out);
}